// PtychoAD_41858751267031
// MI455X (gfx1250) — compile-verified
//
#include <hip/hip_runtime.h>
#include <hip/hip_bf16.h>
#include <math.h>

// ---------------- problem constants (from reference) ----------------
#define O_    2
#define Z_    8
#define OY_   1024
#define OX_   1024
#define P_    8
#define N_    256
#define NPOS_ 512
#define NB_   32
#define NPIX  (N_*N_)          // 65536
#define NIMG  (NB_*O_*P_)      // 512 complex images in the field
#define PI_F  3.14159265358979f

typedef __attribute__((ext_vector_type(2))) float v2f;
typedef __attribute__((ext_vector_type(8))) float v8f;

// ---------------- optional CDNA5 TDM path ----------------
#if defined(__gfx1250__) && __has_builtin(__builtin_amdgcn_tensor_load_to_lds) && __has_builtin(__builtin_amdgcn_s_wait_tensorcnt)
#define USE_TDM 1
#else
#define USE_TDM 0
#endif

#if USE_TDM
typedef __attribute__((ext_vector_type(4))) unsigned tdmv4u;
typedef __attribute__((ext_vector_type(8))) int      tdmv8i;
typedef __attribute__((ext_vector_type(4))) int      tdmv4i;

// Load a 16-column x 256-row f32 slab (row stride 256 floats) from global
// into LDS at byte offset ldsByteOff, packed as [k][16] contiguous rows.
// D# per CDNA5 ISA: group0 = {count=1, lds_addr,
// global_addr, type=2}; group1 = {data_size=4B, dims/strides/tile}.
// 6-operand builtin (clang-23 / therock form):
//   (uint32x4 g0, int32x8 g1, int32x4 g2, int32x4 g3, int32x8 g4, i32 cpol)
__device__ __forceinline__ void tdm_load_col_slab(const float* gsrc, unsigned ldsByteOff) {
  unsigned long long ga = (unsigned long long)(size_t)gsrc;
  tdmv4u g0;
  g0[0] = 1u;                                              // count=1 (valid), user mode
  g0[1] = ldsByteOff;                                      // lds_addr (bytes)
  g0[2] = (unsigned)(ga & 0xFFFFFFFFull);                  // global_addr[31:0]
  g0[3] = (unsigned)((ga >> 32) & 0x01FFFFFFull) | (2u << 30); // addr[56:32] | type=2
  tdmv8i g1;
  g1[0] = (int)(2u << 16);        // workgroup_mask=0, data_size=2 (4 bytes)
  g1[1] = (int)(256u << 16);      // atomic_barrier=0, tensor_dim0[15:0]=256
  g1[2] = (int)(256u << 16);      // tensor_dim0 hi=0, tensor_dim1[15:0]=256
  g1[3] = (int)(16u  << 16);      // tensor_dim1 hi=0, tile_dim0=16
  g1[4] = (int)256;               // tile_dim1=256, tile_dim2=0
  g1[5] = (int)256;               // tensor_dim0_stride[31:0]=256
  g1[6] = (int)(256u << 16);      // stride0 hi=0, tensor_dim1_stride[15:0]=256
  g1[7] = 0;                      // tensor_dim1_stride hi=0
  tdmv4i g2 = {0, 0, 0, 0};
  tdmv4i g3 = {0, 0, 0, 0};
  tdmv8i g4 = {0, 0, 0, 0, 0, 0, 0, 0};
  __builtin_amdgcn_tensor_load_to_lds(g0, g1, g2, g3, g4, 0);
}
#endif

// ---------------- DFT twiddle matrices ----------------
// Wf[k][n] = exp(-2*pi*i*k*n/256); Wi = conj(Wf)/256 (so ifft2 = Wi*X*Wi).
__global__ void ptycho_init_dft(float* wfr, float* wfi, float* wir, float* wii) {
  int i = blockIdx.x * blockDim.x + threadIdx.x;      // 65536 threads
  int k = i >> 8, n = i & 255;
  float ang = -2.0f * PI_F * (float)((k * n) & 255) * (1.0f / 256.0f);
  float s, c;
  __sincosf(ang, &s, &c);
  wfr[i] = c;
  wfi[i] = s;
  wir[i] = c * (1.0f / 256.0f);
  wii[i] = -s * (1.0f / 256.0f);
}

// ---------------- pp output: gather objp crops ----------------
__global__ void ptycho_gather_pp(const float* __restrict__ objp,
                                 const int* __restrict__ crop_pos,
                                 const long long* __restrict__ indices,
                                 float* __restrict__ pp) {
  long long i = (long long)blockIdx.x * blockDim.x + threadIdx.x; // NB*O*Z*N*N
  int x = (int)(i & 255); long long t = i >> 8;
  int y = (int)(t & 255); t >>= 8;
  int z = (int)(t & 7);   t >>= 3;
  int o = (int)(t & 1);   t >>= 1;
  int b = (int)t;
  long long pos = indices[b];
  int cy = crop_pos[2 * pos], cx = crop_pos[2 * pos + 1];
  pp[i] = objp[(((long long)o * Z_ + z) * OY_ + (cy + y)) * OX_ + (cx + x)];
}

// ---------------- probes_k = fft2(probe)[p] * ramp[b] ----------------
__global__ void ptycho_apply_ramp(const float* __restrict__ Fr, const float* __restrict__ Fi,
                                  const float* __restrict__ shifts,
                                  const long long* __restrict__ indices,
                                  float* __restrict__ outR, float* __restrict__ outI) {
  long long i = (long long)blockIdx.x * blockDim.x + threadIdx.x; // NB*P*N*N
  int x = (int)(i & 255); long long t = i >> 8;
  int y = (int)(t & 255); t >>= 8;
  int p = (int)(t & 7);   t >>= 3;
  int b = (int)t;
  long long pos = indices[b];
  float s0 = shifts[2 * pos], s1 = shifts[2 * pos + 1];
  float ang = -2.0f * PI_F * (s0 * (float)y * (1.0f / 256.0f) + s1 * (float)x * (1.0f / 256.0f));
  float sn, c;
  __sincosf(ang, &sn, &c);
  long long fidx = (long long)p * NPIX + (y << 8) + x;
  float fr = Fr[fidx], fi = Fi[fidx];
  outR[i] = fr * c - fi * sn;
  outI[i] = fr * sn + fi * c;
}

// ---------------- psi *= pa * exp(i*pp) for slice z ----------------
__global__ void ptycho_mult_obj(const float* __restrict__ inR, const float* __restrict__ inI,
                                int inBroadcastO,
                                const float* __restrict__ obja,
                                const float* __restrict__ pp,
                                const int* __restrict__ crop_pos,
                                const long long* __restrict__ indices,
                                int z,
                                float* __restrict__ outR, float* __restrict__ outI) {
  long long i = (long long)blockIdx.x * blockDim.x + threadIdx.x; // NB*O*P*N*N
  int x = (int)(i & 255); long long t = i >> 8;
  int y = (int)(t & 255); t >>= 8;
  int p = (int)(t & 7);   t >>= 3;
  int o = (int)(t & 1);   t >>= 1;
  int b = (int)t;
  long long pos = indices[b];
  int cy = crop_pos[2 * pos], cx = crop_pos[2 * pos + 1];
  float pa = obja[(((long long)o * Z_ + z) * OY_ + (cy + y)) * OX_ + (cx + x)];
  float ph = pp[((((long long)b * O_ + o) * Z_ + z) * N_ + y) * N_ + x];
  float s, c;
  __sincosf(ph, &s, &c);
  float obr = pa * c, obi = pa * s;
  long long inIdx = inBroadcastO ? (((long long)b * P_ + p) * NPIX + (y << 8) + x) : i;
  float pr = inR[inIdx], pi = inI[inIdx];
  outR[i] = pr * obr - pi * obi;
  outI[i] = pr * obi + pi * obr;
}

// ---------------- psi_k *= H (in place) ----------------
__global__ void ptycho_mul_H(const float* __restrict__ Hr, const float* __restrict__ Hi,
                             float* __restrict__ R, float* __restrict__ I) {
  long long i = (long long)blockIdx.x * blockDim.x + threadIdx.x; // NIMG*N*N
  int x = (int)(i & 255);
  int y = (int)((i >> 8) & 255);
  float hr = Hr[(y << 8) + x], hi = Hi[(y << 8) + x];
  float r = R[i], im = I[i];
  R[i] = r * hr - im * hi;
  I[i] = r * hi + im * hr;
}

// ---------------- dp = sum_o occu[o] * sum_p |fftshift(psi_k)|^2 ----------------
__global__ void ptycho_accum_dp(const float* __restrict__ R, const float* __restrict__ I,
                                const float* __restrict__ occu,
                                float* __restrict__ dp) {
  long long i = (long long)blockIdx.x * blockDim.x + threadIdx.x; // NB*N*N
  int x = (int)(i & 255); long long t = i >> 8;
  int y = (int)(t & 255);
  int b = (int)(t >> 8);
  int ys = (y + 128) & 255, xs = (x + 128) & 255;
  float acc = 0.0f;
  for (int o = 0; o < O_; ++o) {
    float w = occu[o];
    for (int p = 0; p < P_; ++p) {
      long long idx = ((((long long)b * O_ + o) * P_ + p) * N_ + ys) * N_ + xs;
      float r = R[idx], im = I[idx];
      acc += w * (r * r + im * im);
    }
  }
  dp[i] = acc;
}

// ---------------- batched complex GEMM: C[b] = A[b?] * B[b?], all 256x256 ----------------
// grid = (16 n-tiles, 2 m-groups, nbatch), block = 256 threads (8 wave32).
// Each wave produces one 16x16 complex tile via V_WMMA_F32_16X16X4_F32
// chains (K=256 in steps of 4; RR/II/RI/IR accumulators).
// The B column slab B[:,n0:n0+16] is staged in LDS via the Tensor Data Mover.
__global__ __launch_bounds__(256)
void ptycho_zgemm256(const float* __restrict__ Ar, const float* __restrict__ Ai, long long sA,
                     const float* __restrict__ Br, const float* __restrict__ Bi, long long sB,
                     float* __restrict__ Cr, float* __restrict__ Ci, long long sC) {
  __shared__ float ldsBr[256 * 16];
  __shared__ float ldsBi[256 * 16];
  const int n0 = blockIdx.x * 16;
  const long long batch = blockIdx.z;
  const int wave = threadIdx.x >> 5;
  const int lane = threadIdx.x & 31;
  const int m0 = blockIdx.y * 128 + wave * 16;

  const float* bR = Br + batch * sB;
  const float* bI = Bi + batch * sB;
  const float* aR = Ar + batch * sA;
  const float* aI = Ai + batch * sA;

#if USE_TDM
  if (wave == 0) {                       // wave-uniform branch: one TDM issue per block
    tdm_load_col_slab(bR + n0, (unsigned)(size_t)(void*)ldsBr);
    tdm_load_col_slab(bI + n0, (unsigned)(size_t)(void*)ldsBi);
    __builtin_amdgcn_s_wait_tensorcnt(0);
  }
  __syncthreads();
#else
  for (int i = threadIdx.x; i < 256 * 16; i += 256) {
    int k = i >> 4, n = i & 15;
    ldsBr[i] = bR[(long long)k * 256 + n0 + n];
    ldsBi[i] = bI[(long long)k * 256 + n0 + n];
  }
  __syncthreads();
#endif

  v8f accRR = {}, accII = {}, accRI = {}, accIR = {};

  // wave32 WMMA operand layout (CDNA5 ISA):
  //  A 16x4 f32: lane L -> M = L&15; VGPR j -> K = kk + j + 2*(L>=16)
  //  B  4x16 f32: lane L -> N = L&15; VGPR j -> K = kk + j + 2*(L>=16)
  const int mRow = m0 + (lane & 15);
  const int kOff = (lane >> 4) << 1;
  const int nCol = lane & 15;

  const float* aRp = aR + (long long)mRow * 256 + kOff;   // advances by K
  const float* aIp = aI + (long long)mRow * 256 + kOff;
  const float* bRp = ldsBr + kOff * 16 + nCol;            // advances by K*16
  const float* bIp = ldsBi + kOff * 16 + nCol;

  // Unrolled by 2: per body 4 global b64 loads + 4 LDS 2addr loads + 8 WMMA,
  // plus one unconditional speculative prefetch pair 64B ahead (ISA 10.5:
  // invalid/overrun speculative prefetches are silently dropped).
  for (int kk = 0; kk < 256; kk += 8) {
    __builtin_prefetch(aRp + 16, 0, 3);
    __builtin_prefetch(aIp + 16, 0, 3);

    v2f ar0 = *(const v2f*)(aRp);
    v2f ai0 = *(const v2f*)(aIp);
    v2f br0; br0[0] = bRp[0];      br0[1] = bRp[16];
    v2f bi0; bi0[0] = bIp[0];      bi0[1] = bIp[16];
    accRR = __builtin_amdgcn_wmma_f32_16x16x4_f32(false, ar0, false, br0, (short)0, accRR, false, false);
    accII = __builtin_amdgcn_wmma_f32_16x16x4_f32(false, ai0, false, bi0, (short)0, accII, false, false);
    accRI = __builtin_amdgcn_wmma_f32_16x16x4_f32(false, ar0, false, bi0, (short)0, accRI, false, false);
    accIR = __builtin_amdgcn_wmma_f32_16x16x4_f32(false, ai0, false, br0, (short)0, accIR, false, false);

    v2f ar1 = *(const v2f*)(aRp + 4);
    v2f ai1 = *(const v2f*)(aIp + 4);
    v2f br1; br1[0] = bRp[64];     br1[1] = bRp[80];
    v2f bi1; bi1[0] = bIp[64];     bi1[1] = bIp[80];
    accRR = __builtin_amdgcn_wmma_f32_16x16x4_f32(false, ar1, false, br1, (short)0, accRR, false, false);
    accII = __builtin_amdgcn_wmma_f32_16x16x4_f32(false, ai1, false, bi1, (short)0, accII, false, false);
    accRI = __builtin_amdgcn_wmma_f32_16x16x4_f32(false, ar1, false, bi1, (short)0, accRI, false, false);
    accIR = __builtin_amdgcn_wmma_f32_16x16x4_f32(false, ai1, false, br1, (short)0, accIR, false, false);

    aRp += 8; aIp += 8;
    bRp += 128; bIp += 128;
  }

  float* cR = Cr + batch * sC;
  float* cI = Ci + batch * sC;
  // C layout: VGPR r, lane L -> M = r + 8*(L>=16), N = L&15
  for (int r = 0; r < 8; ++r) {
    int mm = m0 + r + ((lane >> 4) << 3);
    long long cidx = (long long)mm * 256 + n0 + (lane & 15);
    cR[cidx] = accRR[r] - accII[r];
    cI[cidx] = accRI[r] + accIR[r];
  }
}

// ---------------- host-side orchestration ----------------
static inline void launch_fft2(const float* Wr, const float* Wi,
                               float* Xr, float* Xi, float* Tr, float* Ti,
                               int nbatch, hipStream_t stream) {
  // X <- W * X * W   (two GEMM passes, T is scratch)
  ptycho_zgemm256<<<dim3(16, 2, nbatch), 256, 0, stream>>>(Wr, Wi, 0LL, Xr, Xi, (long long)NPIX, Tr, Ti, (long long)NPIX);
  ptycho_zgemm256<<<dim3(16, 2, nbatch), 256, 0, stream>>>(Tr, Ti, (long long)NPIX, Wr, Wi, 0LL, Xr, Xi, (long long)NPIX);
}

extern "C" void kernel_launch(void* const* d_in, const int* in_sizes, int n_in,
                              void* d_out, int out_size, void* d_ws, size_t ws_size,
                              hipStream_t stream) {
  (void)in_sizes; (void)n_in; (void)out_size; (void)ws_size;

  const float*     obja    = (const float*)d_in[0];
  const float*     objp    = (const float*)d_in[1];
  const float*     probe_r = (const float*)d_in[2];
  const float*     probe_i = (const float*)d_in[3];
  const float*     H_r     = (const float*)d_in[4];
  const float*     H_i     = (const float*)d_in[5];
  const float*     occu    = (const float*)d_in[6];
  const float*     shifts  = (const float*)d_in[7];
  const int*       crop    = (const int*)d_in[8];
  const long long* idx     = (const long long*)d_in[9];

  float* dp = (float*)d_out;                       // (NB, N, N)
  float* pp = dp + (size_t)NB_ * NPIX;             // (NB, O, Z, N, N)

  // workspace carve-up (f32 planes)
  float* p   = (float*)d_ws;
  float* Wfr = p; p += NPIX;
  float* Wfi = p; p += NPIX;
  float* Wir = p; p += NPIX;
  float* Wii = p; p += NPIX;
  float* Fr  = p; p += (size_t)P_ * NPIX;          // fft2(probe)
  float* Fi  = p; p += (size_t)P_ * NPIX;
  float* PRr = p; p += (size_t)NB_ * P_ * NPIX;    // shifted probes
  float* PRi = p; p += (size_t)NB_ * P_ * NPIX;
  float* Xr  = p; p += (size_t)NIMG * NPIX;        // psi (ping)
  float* Xi  = p; p += (size_t)NIMG * NPIX;
  float* Tr  = p; p += (size_t)NIMG * NPIX;        // scratch (pong)
  float* Ti  = p; p += (size_t)NIMG * NPIX;

  const int EW = 256;                              // elementwise block size
  const long long totPP   = (long long)NB_ * O_ * Z_ * NPIX;   // 33.5M
  const long long totPsi  = (long long)NB_ * O_ * P_ * NPIX;   // 33.5M
  const long long totRamp = (long long)NB_ * P_ * NPIX;        // 16.8M
  const long long totDP   = (long long)NB_ * NPIX;             // 2.1M

  // 0) DFT matrices + pp output
  ptycho_init_dft<<<NPIX / EW, EW, 0, stream>>>(Wfr, Wfi, Wir, Wii);
  ptycho_gather_pp<<<(unsigned)(totPP / EW), EW, 0, stream>>>(objp, crop, idx, pp);

  // 1) F = fft2(probe)  (8 images, probe inputs used directly as B operand)
  ptycho_zgemm256<<<dim3(16, 2, P_), 256, 0, stream>>>(Wfr, Wfi, 0LL, probe_r, probe_i, (long long)NPIX, Tr, Ti, (long long)NPIX);
  ptycho_zgemm256<<<dim3(16, 2, P_), 256, 0, stream>>>(Tr, Ti, (long long)NPIX, Wfr, Wfi, 0LL, Fr, Fi, (long long)NPIX);

  // 2) probes = ifft2(F * ramp[b])   (NB*P = 256 images)
  ptycho_apply_ramp<<<(unsigned)(totRamp / EW), EW, 0, stream>>>(Fr, Fi, shifts, idx, Tr, Ti);
  ptycho_zgemm256<<<dim3(16, 2, NB_ * P_), 256, 0, stream>>>(Wir, Wii, 0LL, Tr, Ti, (long long)NPIX, Xr, Xi, (long long)NPIX);
  ptycho_zgemm256<<<dim3(16, 2, NB_ * P_), 256, 0, stream>>>(Xr, Xi, (long long)NPIX, Wir, Wii, 0LL, PRr, PRi, (long long)NPIX);

  // 3) psi = probes (broadcast over O) * obj[z=0]
  ptycho_mult_obj<<<(unsigned)(totPsi / EW), EW, 0, stream>>>(PRr, PRi, 1, obja, pp, crop, idx, 0, Xr, Xi);

  // 4) slices 1..7: propagate then multiply
  for (int z = 1; z < Z_; ++z) {
    launch_fft2(Wfr, Wfi, Xr, Xi, Tr, Ti, NIMG, stream);                         // fft2
    ptycho_mul_H<<<(unsigned)(totPsi / EW), EW, 0, stream>>>(H_r, H_i, Xr, Xi);  // * H
    launch_fft2(Wir, Wii, Xr, Xi, Tr, Ti, NIMG, stream);                         // ifft2
    ptycho_mult_obj<<<(unsigned)(totPsi / EW), EW, 0, stream>>>(Xr, Xi, 0, obja, pp, crop, idx, z, Xr, Xi);
  }

  // 5) psi_k = fft2(psi); dp = sum occu * |fftshift(psi_k)|^2
  launch_fft2(Wfr, Wfi, Xr, Xi, Tr, Ti, NIMG, stream);
  ptycho_accum_dp<<<(unsigned)(totDP / EW), EW, 0, stream>>>(Xr, Xi, occu, dp);
}